// MixedHOMVector_86251533238416
// MI455X (gfx1250) — compile-verified
//
#include <hip/hip_runtime.h>

typedef float v2f __attribute__((ext_vector_type(2)));
typedef float v4f __attribute__((ext_vector_type(4)));
typedef float v8f __attribute__((ext_vector_type(8)));

constexpr int B = 32;
constexpr int T = 8192;
constexpr int C = 256;
constexpr int TCHUNK = 256;
constexpr int NCHUNK = T / TCHUNK;   // 32 partial chunks along time
constexpr float EPS = 1e-6f;

// ---------------------------------------------------------------------------
// Pass 1: streaming raw-moment partial sums, b128-vectorized.
// Grid: (NCHUNK, B) blocks x 256 threads.
// Thread (cx = tid%64, ty = tid/64) loads float4 over channels 4cx..4cx+3,
// rows t = ty, ty+4, ... (global_load_b128, non-temporal, fully coalesced).
// The 4 row-phase partials are combined in LDS, then written to
// ws[b][chunk][m][c]  (m: 0=sum x^p, 1=sum x, 2=sum x^2, 3=sum x^3, 4=sum x^4)
// ---------------------------------------------------------------------------
__global__ __launch_bounds__(256) void pass1_moments(const float* __restrict__ x,
                                                     const float* __restrict__ pptr,
                                                     float* __restrict__ ws) {
  __shared__ v4f red[4][5][C / 4];   // 4 phases x 5 moments x 64 groups = 20 KB

  const int tid   = threadIdx.x;
  const int cx    = tid & 63;        // channel group (4 channels each)
  const int ty    = tid >> 6;        // row phase 0..3
  const int chunk = blockIdx.x;
  const int b     = blockIdx.y;
  const float p   = pptr[0];

  const v4f* __restrict__ base =
      (const v4f*)(x + ((size_t)b * T + (size_t)chunk * TCHUNK) * C) + cx;

  float s1[4] = {}, s2[4] = {}, s3[4] = {}, s4[4] = {}, sp[4] = {};

  if (p == 1.0f) {
    // Fast path: x^p == x -> pure HBM-bound stream, no transcendentals.
#pragma unroll 4
    for (int t = ty; t < TCHUNK; t += 4) {
      v4f v = __builtin_nontemporal_load(base + (size_t)t * (C / 4));
#pragma unroll
      for (int j = 0; j < 4; ++j) {
        float vj = v[j];
        float v2 = vj * vj;
        s1[j] += vj;
        s2[j] += v2;
        s3[j] += v2 * vj;
        s4[j] += v2 * v2;
      }
    }
#pragma unroll
    for (int j = 0; j < 4; ++j) sp[j] = s1[j];
  } else {
    // Generic path: x^p = exp2(p * log2(x)); x in [0,1), p > 0 handled.
#pragma unroll 2
    for (int t = ty; t < TCHUNK; t += 4) {
      v4f v = __builtin_nontemporal_load(base + (size_t)t * (C / 4));
#pragma unroll
      for (int j = 0; j < 4; ++j) {
        float vj = v[j];
        float v2 = vj * vj;
        s1[j] += vj;
        s2[j] += v2;
        s3[j] += v2 * vj;
        s4[j] += v2 * v2;
        sp[j] += __builtin_amdgcn_exp2f(p * __builtin_amdgcn_logf(vj));
      }
    }
  }

  // Stage per-phase partials in LDS (b128 stores, conflict-free).
  v4f t0, t1, t2, t3, t4;
#pragma unroll
  for (int j = 0; j < 4; ++j) { t0[j] = sp[j]; t1[j] = s1[j]; t2[j] = s2[j]; t3[j] = s3[j]; t4[j] = s4[j]; }
  red[ty][0][cx] = t0;
  red[ty][1][cx] = t1;
  red[ty][2][cx] = t2;
  red[ty][3][cx] = t3;
  red[ty][4][cx] = t4;
  __syncthreads();

  // Combine the 4 row phases: thread == channel, coalesced ws writes.
  const float* __restrict__ rf = (const float*)red;
  float* __restrict__ o = ws + (((size_t)b * NCHUNK + chunk) * 5) * C + tid;
#pragma unroll
  for (int m = 0; m < 5; ++m) {
    float s = rf[(0 * 5 + m) * C + tid] + rf[(1 * 5 + m) * C + tid] +
              rf[(2 * 5 + m) * C + tid] + rf[(3 * 5 + m) * C + tid];
    o[(size_t)m * C] = s;
  }
}

// ---------------------------------------------------------------------------
// Pass 2: reduce NCHUNK partials per (b, c) with V_WMMA_F32_16X16X4_F32
// (A = all-ones 16x4, B = 4 chunk-rows x 16 channels), then finalize stats.
// One full wave per (b, 16-channel tile): 32 * 16 = 512 waves exactly,
// launched as 64 blocks x 256 threads -> EXEC all ones for every WMMA.
//
// B-operand layout (4x16 f32, 2 VGPRs):
//   VGPR0: lanes 0-15 = row K=0 (N=lane), lanes 16-31 = row K=2 (N=lane-16)
//   VGPR1: lanes 0-15 = row K=1,          lanes 16-31 = row K=3
// D rows are all identical (A==1), so acc[0] in every lane = colsum[lane%16].
// ---------------------------------------------------------------------------
__global__ __launch_bounds__(256) void pass2_wmma_finalize(const float* __restrict__ ws,
                                                           float* __restrict__ out) {
  const int tid   = blockIdx.x * 256 + threadIdx.x;
  const int wid   = tid >> 5;      // 0..511
  const int lane  = tid & 31;
  const int b     = wid >> 4;      // 0..31
  const int ctile = wid & 15;      // 0..15 (16 channels each)
  const int n     = lane & 15;
  const int khalf = lane >> 4;     // 0 or 1 (selects K rows 0/1 vs 2/3)
  const int cg    = ctile * 16 + n;

  v2f ones;
  ones.x = 1.0f;
  ones.y = 1.0f;

  v8f acc[5] = {};

  const float* __restrict__ wb = ws + (size_t)b * NCHUNK * 5 * C;

#pragma unroll
  for (int m = 0; m < 5; ++m) {
#pragma unroll
    for (int g = 0; g < NCHUNK / 4; ++g) {          // 8 groups of 4 chunks
      const size_t row0 = (size_t)(4 * g + 2 * khalf);
      v2f bm;
      bm.x = wb[(row0 * 5 + m) * C + cg];           // K = 2*khalf
      bm.y = wb[((row0 + 1) * 5 + m) * C + cg];     // K = 2*khalf + 1
      acc[m] = __builtin_amdgcn_wmma_f32_16x16x4_f32(
          /*neg_a=*/false, ones, /*neg_b=*/false, bm,
          /*c_mod=*/(short)0, acc[m], /*reuse_a=*/false, /*reuse_b=*/false);
    }
  }

  const float invT = 1.0f / (float)T;
  const float Sp = acc[0][0];
  const float S1 = acc[1][0];
  const float S2 = acc[2][0];
  const float S3 = acc[3][0];
  const float S4 = acc[4][0];

  const float grp = Sp * invT;            // mean(x^p)
  const float mu  = S1 * invT;
  const float ex2 = S2 * invT;
  const float ex3 = S3 * invT;
  const float ex4 = S4 * invT;

  const float mu2 = mu * mu;
  const float var = ex2 - mu2;
  const float m3  = ex3 - 3.0f * mu * ex2 + 2.0f * mu * mu2;
  const float m4  = ex4 - 4.0f * mu * ex3 + 6.0f * mu2 * ex2 - 3.0f * mu2 * mu2;

  const float sd   = sqrtf(var + EPS);
  const float sd2  = sd * sd;
  const float skew = m3 / (sd2 * sd);
  const float kurt = m4 / (sd2 * sd2);

  if (lane < 16) {                        // lanes 16-31 hold duplicate channels
    float* __restrict__ o = out + (size_t)b * (4 * C) + cg;
    o[0 * C] = grp;
    o[1 * C] = var;
    o[2 * C] = skew;
    o[3 * C] = kurt;
  }
}

// ---------------------------------------------------------------------------
extern "C" void kernel_launch(void* const* d_in, const int* in_sizes, int n_in,
                              void* d_out, int out_size, void* d_ws, size_t ws_size,
                              hipStream_t stream) {
  (void)in_sizes; (void)n_in; (void)out_size; (void)ws_size;

  const float* x = (const float*)d_in[0];
  const float* p = (const float*)d_in[1];
  float* out = (float*)d_out;
  float* ws  = (float*)d_ws;   // needs B*NCHUNK*5*C*4 = 5 MiB

  dim3 g1(NCHUNK, B);
  pass1_moments<<<g1, dim3(256), 0, stream>>>(x, p, ws);

  // 512 waves total == B * (C/16); 64 blocks x 256 threads, all waves full.
  pass2_wmma_finalize<<<64, 256, 0, stream>>>(ws, out);
}